// LogTransform_44779329028117
// MI455X (gfx1250) — compile-verified
//
#include <hip/hip_runtime.h>

// Batched SPD matrix log via inverse scaling-and-squaring + Newton-Schulz
// square roots: pure 64x64x64 fp32 GEMMs -> V_WMMA_F32_16X16X4_F32.
//
// Key trick: every operand matrix in this algorithm (Y, Z, T, W, G, P1, ...)
// is a polynomial of the same SPD input X, hence symmetric. So the B operand
// is read as its transpose (B[k,col] = B[col,k]): each lane walks a row, the
// two K-adjacent elements are contiguous dwords, and every fragment load is a
// single ds_load_2addr_b32 into consecutive VGPRs (no repack movs).

typedef __attribute__((ext_vector_type(2))) float v2f;
typedef __attribute__((ext_vector_type(8))) float v8f;

#define NS_ITERS 10   // Newton-Schulz iterations per square root
#define M_SQRTS  5    // number of repeated square roots (scaling depth)

// Store epilogue modes (fused diagonal shifts)
#define EPI_PLAIN   0  // C
#define EPI_NS_T    1  // 1.5*I - 0.5*C      (Newton-Schulz T update)
#define EPI_MHALF_I 2  // C - 0.5*I          (Mercator step)
#define EPI_PLUS_I  3  // C + I              (Mercator step)

// C = A * B for 64x64 f32 symmetric matrices in LDS arena S.
// Wave w owns row band tr=(w>>1)*16 and columns tc0=(w&1)*32, tc1=tc0+16.
template<int MODE>
__device__ __forceinline__ void gemm64(float* S, int co, int ao, int bo,
                                       int wave, int lane)
{
    const int hi  = lane >> 4;      // 0: lanes 0-15, 1: lanes 16-31
    const int lo  = lane & 15;
    const int tr  = (wave >> 1) * 16;
    const int tc0 = (wave & 1) * 32;
    const int tc1 = tc0 + 16;

    // Per-lane row bases: A row tr+lo; B read as transpose -> rows tc+lo
    const int ar  = ao + (tr + lo) * 64;
    const int br0 = bo + (tc0 + lo) * 64;
    const int br1 = bo + (tc1 + lo) * 64;

    v8f acc0 = {0.f, 0.f, 0.f, 0.f, 0.f, 0.f, 0.f, 0.f};
    v8f acc1 = {0.f, 0.f, 0.f, 0.f, 0.f, 0.f, 0.f, 0.f};
#pragma unroll
    for (int k0 = 0; k0 < 64; k0 += 4) {
        const int ka = k0 + hi * 2;
        v2f a, b0, b1;
        // A 16x4 frag (shared by both tiles): K = ka, ka+1 (adjacent dwords)
        a.x  = S[ar + ka];
        a.y  = S[ar + ka + 1];
        // B 4x16 frags via symmetry: B[ka, col] = B[col, ka] (adjacent dwords)
        b0.x = S[br0 + ka];
        b0.y = S[br0 + ka + 1];
        b1.x = S[br1 + ka];
        b1.y = S[br1 + ka + 1];
        acc0 = __builtin_amdgcn_wmma_f32_16x16x4_f32(
            false, a, false, b0, (short)0, acc0, false, false);
        acc1 = __builtin_amdgcn_wmma_f32_16x16x4_f32(
            false, a, false, b1, (short)0, acc1, false, false);
    }
    // C/D layout: VGPR v -> row m = v + 8*hi, col = lo
#pragma unroll
    for (int v = 0; v < 8; ++v) {
        const int row = tr + hi * 8 + v;
        {
            const bool dg = (row == tc0 + lo);
            float r = acc0[v];
            if (MODE == EPI_NS_T)    r = (dg ? 1.5f : 0.f) - 0.5f * r;
            if (MODE == EPI_MHALF_I) r = r - (dg ? 0.5f : 0.f);
            if (MODE == EPI_PLUS_I)  r = r + (dg ? 1.f : 0.f);
            S[co + row * 64 + tc0 + lo] = r;
        }
        {
            const bool dg = (row == tc1 + lo);
            float r = acc1[v];
            if (MODE == EPI_NS_T)    r = (dg ? 1.5f : 0.f) - 0.5f * r;
            if (MODE == EPI_MHALF_I) r = r - (dg ? 0.5f : 0.f);
            if (MODE == EPI_PLUS_I)  r = r + (dg ? 1.f : 0.f);
            S[co + row * 64 + tc1 + lo] = r;
        }
    }
}

// Final GEMM: out = sc * (A*B) + logc * I, written straight to global.
__device__ __forceinline__ void gemm64_out(float* __restrict__ og,
                                           const float* S, int ao, int bo,
                                           float sc, float logc,
                                           int wave, int lane)
{
    const int hi  = lane >> 4;
    const int lo  = lane & 15;
    const int tr  = (wave >> 1) * 16;
    const int tc0 = (wave & 1) * 32;
    const int tc1 = tc0 + 16;

    const int ar  = ao + (tr + lo) * 64;
    const int br0 = bo + (tc0 + lo) * 64;
    const int br1 = bo + (tc1 + lo) * 64;

    v8f acc0 = {0.f, 0.f, 0.f, 0.f, 0.f, 0.f, 0.f, 0.f};
    v8f acc1 = {0.f, 0.f, 0.f, 0.f, 0.f, 0.f, 0.f, 0.f};
#pragma unroll
    for (int k0 = 0; k0 < 64; k0 += 4) {
        const int ka = k0 + hi * 2;
        v2f a, b0, b1;
        a.x  = S[ar + ka];
        a.y  = S[ar + ka + 1];
        b0.x = S[br0 + ka];
        b0.y = S[br0 + ka + 1];
        b1.x = S[br1 + ka];
        b1.y = S[br1 + ka + 1];
        acc0 = __builtin_amdgcn_wmma_f32_16x16x4_f32(
            false, a, false, b0, (short)0, acc0, false, false);
        acc1 = __builtin_amdgcn_wmma_f32_16x16x4_f32(
            false, a, false, b1, (short)0, acc1, false, false);
    }
#pragma unroll
    for (int v = 0; v < 8; ++v) {
        const int row = tr + hi * 8 + v;
        og[row * 64 + tc0 + lo] =
            sc * acc0[v] + ((row == tc0 + lo) ? logc : 0.f);
        og[row * 64 + tc1 + lo] =
            sc * acc1[v] + ((row == tc1 + lo) ? logc : 0.f);
    }
}

__global__ __launch_bounds__(256)
void logm_spd64_kernel(const float* __restrict__ x, float* __restrict__ out)
{
    // 4 x 64x64 f32 panels: Y, Z, T, W (rotated by integer offset)
    __shared__ __align__(16) float S[4 * 4096];

    const int tid  = threadIdx.x;
    const int wave = tid >> 5;
    const int lane = tid & 31;
    const size_t b = blockIdx.x;

    int oy = 0, oz = 4096, ot = 8192, ow = 12288;

    // ---- load X into Y panel (vectorized, 4 float4 per thread) ----
    {
        const float4* src = (const float4*)(x + b * 4096);
        float4* dst = (float4*)(S + oy);
#pragma unroll
        for (int i = 0; i < 4; ++i)
            dst[tid + 256 * i] = src[tid + 256 * i];
    }
    __syncthreads();

    // ---- trace normalization: c = tr(X) (sum of eigenvalues) ----
    if (tid < 64) S[ow + tid] = S[oy + tid * 65];   // W panel as scratch
    __syncthreads();
    float c = 0.f;
#pragma unroll 1
    for (int i = 0; i < 64; ++i) c += S[ow + i];
    c = fmaxf(c, 1e-20f);
    const float logc  = logf(c);
    const float inv_c = 1.0f / c;

    // Y = X / c   (eigenvalues now in (0, 1])
#pragma unroll
    for (int i = 0; i < 16; ++i) {
        const int idx = tid * 16 + i;
        S[oy + idx] *= inv_c;
    }

    // ---- m repeated square roots via coupled Newton-Schulz ----
#pragma unroll 1
    for (int s = 0; s < M_SQRTS; ++s) {
        // Z = I
#pragma unroll
        for (int i = 0; i < 16; ++i) {
            const int idx = tid * 16 + i;
            S[oz + idx] = ((idx >> 6) == (idx & 63)) ? 1.f : 0.f;
        }
        __syncthreads();

#pragma unroll 1
        for (int it = 0; it < NS_ITERS; ++it) {
            gemm64<EPI_NS_T>(S, ot, oz, oy, wave, lane); // T = 1.5I-0.5*Z*Y
            __syncthreads();
            gemm64<EPI_PLAIN>(S, ow, oy, ot, wave, lane); // W = Y*T (new Y)
            __syncthreads();
            gemm64<EPI_PLAIN>(S, oy, ot, oz, wave, lane); // Ybuf = T*Z (new Z)
            __syncthreads();
            const int tmp = oy; oy = ow; ow = oz; oz = tmp;
        }
        // Y panel now holds sqrt of previous Y
    }

    // ---- R = Y ~ (X/c)^(1/2^m), Mercator series for log R ----
    // G -> T ;  P1 = (1/3)I - (1/4)G -> Z
#pragma unroll
    for (int i = 0; i < 16; ++i) {
        const int idx = tid * 16 + i;
        const bool dg = ((idx >> 6) == (idx & 63));
        const float g = S[oy + idx] - (dg ? 1.f : 0.f);
        S[ot + idx] = g;
        S[oz + idx] = (dg ? (1.f / 3.f) : 0.f) - 0.25f * g;
    }
    __syncthreads();
    gemm64<EPI_MHALF_I>(S, ow, ot, oz, wave, lane);  // W = G*P1 - 0.5I
    __syncthreads();
    gemm64<EPI_PLUS_I>(S, oz, ot, ow, wave, lane);   // Z = G*W + I
    __syncthreads();

    // ---- out = 2^m * (G*Z) + log(c) * I, straight to global ----
    const float sc = (float)(1 << M_SQRTS);
    gemm64_out(out + b * 4096, S, ot, oz, sc, logc, wave, lane);
}

extern "C" void kernel_launch(void* const* d_in, const int* in_sizes, int n_in,
                              void* d_out, int out_size, void* d_ws, size_t ws_size,
                              hipStream_t stream)
{
    const float* x = (const float*)d_in[0];
    float* out = (float*)d_out;
    const int nb = in_sizes[0] / 4096;   // 8192 matrices, one workgroup each
    logm_spd64_kernel<<<dim3(nb), dim3(256), 0, stream>>>(x, out);
}